// SAGE_6416681140927
// MI455X (gfx1250) — compile-verified
//
#include <hip/hip_runtime.h>

typedef __attribute__((ext_vector_type(2))) float v2f;
typedef __attribute__((ext_vector_type(8))) float v8f;

#define NEG_SLOPE 0.01f

// D(16x16,f32) = A(16x4,f32) x B(4x16,f32) + C   -- full-precision CDNA5 WMMA
__device__ __forceinline__ v8f wmma_f32_k4(v2f a, v2f b, v8f c) {
  return __builtin_amdgcn_wmma_f32_16x16x4_f32(
      /*neg_a=*/false, a, /*neg_b=*/false, b,
      /*c_mod=*/(short)0, c, /*reuse_a=*/false, /*reuse_b=*/false);
}

// Intra-wave LDS RAW/WAR fence: CDNA5 split DS counter + compiler barrier.
__device__ __forceinline__ void lds_fence() {
  asm volatile("s_wait_dscnt 0" ::: "memory");
}

__global__ void zero_f32(float* __restrict__ p, long long n) {
  long long i = (long long)blockIdx.x * blockDim.x + threadIdx.x;
  long long stride = (long long)gridDim.x * blockDim.x;
  for (; i < n; i += stride) p[i] = 0.0f;
}

// 4 lanes per edge; each lane moves one float4 (wave = 8 edges, fully coalesced
// 64B row reads) and issues 4 non-returning global f32 atomics into L2-resident agg.
__global__ void edge_scatter(const float* __restrict__ x,
                             const int* __restrict__ esrc,
                             const int* __restrict__ edst,
                             float* __restrict__ agg, long long n_edges) {
  long long t = (long long)blockIdx.x * blockDim.x + threadIdx.x;
  const int q = (int)(t & 3);            // which float4 quarter of the 16-wide row
  long long e = t >> 2;
  long long stride = ((long long)gridDim.x * blockDim.x) >> 2;
  for (; e < n_edges; e += stride) {
    const int s = esrc[e];
    const int d = edst[e];
    const float4 v = ((const float4*)(x + (long long)s * 16))[q];
    float* ap = agg + (long long)d * 16 + q * 4;
    atomicAdd(ap + 0, v.x);
    atomicAdd(ap + 1, v.y);
    atomicAdd(ap + 2, v.z);
    atomicAdd(ap + 3, v.w);
  }
}

// Shared 3-layer tile compute. aA/aX are the preloaded layer-0 A fragments
// (agg tile and x tile). All WMMAs run with EXEC all-1s (callers guarantee
// wave-uniform control flow here).
__device__ __forceinline__ void mlp_tile(
    const v2f aA[4], const v2f aX[4],
    const float* __restrict__ Wl, const float* __restrict__ bl,
    const float* __restrict__ Wr,
    const float* __restrict__ W1, const float* __restrict__ b1,
    const float* __restrict__ W2, const float* __restrict__ b2,
    float (*Sw)[18], float* __restrict__ out,
    int r0, int batch, int lane16, int half, int ko) {
  const int wrow = lane16 * 16 + ko;     // B[k][n] = W[n][k]; lane = column n

  // Preload all B fragments so loads clause under a single wait.
  v2f bL[4], bR[4];
#pragma unroll
  for (int i = 0; i < 4; ++i) {
    bL[i] = *(const v2f*)(Wl + wrow + 4 * i);
    bR[i] = *(const v2f*)(Wr + wrow + 4 * i);
  }

  // ---- layer 0: acc = agg_tile @ Wl^T + x_tile @ Wr^T ----
  v8f acc = {};
#pragma unroll
  for (int i = 0; i < 4; ++i) acc = wmma_f32_k4(aA[i], bL[i], acc);
#pragma unroll
  for (int i = 0; i < 4; ++i) acc = wmma_f32_k4(aX[i], bR[i], acc);
  {
    const float bias = bl[lane16];       // C/D layout: N = lane16 for all regs
#pragma unroll
    for (int r = 0; r < 8; ++r) {
      float v = acc[r] + bias;
      v = (v >= 0.0f) ? v : NEG_SLOPE * v;
      Sw[r + 8 * half][lane16] = v;      // C/D row = r + 8*half
    }
  }
  lds_fence();

  // ---- layer 1: h1 = leaky(h @ W1^T + b1) ----
  v2f a1[4], b1f[4];
#pragma unroll
  for (int i = 0; i < 4; ++i) {
    a1[i]  = *(const v2f*)(&Sw[lane16][4 * i + ko]);   // A[m=lane16][k]
    b1f[i] = *(const v2f*)(W1 + wrow + 4 * i);
  }
  v8f acc1 = {};
#pragma unroll
  for (int i = 0; i < 4; ++i) acc1 = wmma_f32_k4(a1[i], b1f[i], acc1);
  lds_fence();                           // WAR: reads drained before overwrite
  {
    const float bias = b1[lane16];
#pragma unroll
    for (int r = 0; r < 8; ++r) {
      float v = acc1[r] + bias;
      v = (v >= 0.0f) ? v : NEG_SLOPE * v;
      Sw[r + 8 * half][lane16] = v;
    }
  }
  lds_fence();

  // ---- layer 2: out = h1 @ W2^T + b2 ----
  v2f a2[4], b2f[4];
#pragma unroll
  for (int i = 0; i < 4; ++i) {
    a2[i]  = *(const v2f*)(&Sw[lane16][4 * i + ko]);
    b2f[i] = *(const v2f*)(W2 + wrow + 4 * i);
  }
  v8f acc2 = {};
#pragma unroll
  for (int i = 0; i < 4; ++i) acc2 = wmma_f32_k4(a2[i], b2f[i], acc2);
  {
    const float bias = b2[lane16];
#pragma unroll
    for (int r = 0; r < 8; ++r) {
      const int orow = r0 + r + 8 * half;
      if (orow < batch)
        out[(long long)orow * 16 + lane16] = acc2[r] + bias;
    }
  }
}

// One wave32 per 16-row tile; three 16x16 f32 GEMM layers via v_wmma_f32_16x16x4_f32.
__global__ void __launch_bounds__(128)
sage_mlp_wmma(const float* __restrict__ agg, const float* __restrict__ x,
              const float* __restrict__ Wl, const float* __restrict__ bl,
              const float* __restrict__ Wr,
              const float* __restrict__ W1, const float* __restrict__ b1,
              const float* __restrict__ W2, const float* __restrict__ b2,
              float* __restrict__ out, int batch) {
  __shared__ float S[4][16][18];         // pad 18: conflict-free, 8B-aligned rows
  const int wave   = threadIdx.x >> 5;
  const int l      = threadIdx.x & 31;
  const int half   = l >> 4;             // lanes 0-15: K pair {0,1}; 16-31: {2,3}
  const int lane16 = l & 15;
  const int ko     = 2 * half;
  const int tile   = blockIdx.x * 4 + wave;
  if (tile * 16 >= batch) return;        // wave-uniform: EXEC all-1s for WMMA
  const int r0 = tile * 16;
  float (*Sw)[18] = S[wave];

  v2f aA[4], aX[4];
  if (r0 + 16 <= batch) {
    // Fast path (the only one taken when batch % 16 == 0): unpredicated clause
    // of 8 global_load_b64 covering both layer-0 A tiles.
    const long long rb = (long long)(r0 + lane16) * 16 + ko;
#pragma unroll
    for (int i = 0; i < 4; ++i) {
      aA[i] = *(const v2f*)(agg + rb + 4 * i);
      aX[i] = *(const v2f*)(x + rb + 4 * i);
    }
  } else {
    // Tail tile: stage zero-padded rows through LDS (divergence confined here;
    // control flow reconverges before any WMMA).
    const int row = r0 + lane16;
    const bool rowv = row < batch;
#pragma unroll
    for (int c = 0; c < 8; ++c)
      Sw[lane16][half * 8 + c] = rowv ? agg[(long long)row * 16 + half * 8 + c] : 0.0f;
    lds_fence();
#pragma unroll
    for (int i = 0; i < 4; ++i) aA[i] = *(const v2f*)(&Sw[lane16][4 * i + ko]);
    lds_fence();
#pragma unroll
    for (int c = 0; c < 8; ++c)
      Sw[lane16][half * 8 + c] = rowv ? x[(long long)row * 16 + half * 8 + c] : 0.0f;
    lds_fence();
#pragma unroll
    for (int i = 0; i < 4; ++i) aX[i] = *(const v2f*)(&Sw[lane16][4 * i + ko]);
    lds_fence();
  }

  mlp_tile(aA, aX, Wl, bl, Wr, W1, b1, W2, b2, Sw, out,
           r0, batch, lane16, half, ko);
}

extern "C" void kernel_launch(void* const* d_in, const int* in_sizes, int n_in,
                              void* d_out, int out_size, void* d_ws, size_t ws_size,
                              hipStream_t stream) {
  const float* x  = (const float*)d_in[0];
  const float* Wl = (const float*)d_in[1];
  const float* bl = (const float*)d_in[2];
  const float* Wr = (const float*)d_in[3];
  const float* W1 = (const float*)d_in[4];
  const float* b1 = (const float*)d_in[5];
  const float* W2 = (const float*)d_in[6];
  const float* b2 = (const float*)d_in[7];
  const int* esrc = (const int*)d_in[8];
  const int* edst = (const int*)d_in[9];

  const long long n_edges = in_sizes[8];
  const int batch = out_size / 16;       // D = 16
  float* agg = (float*)d_ws;             // batch*16 f32 accumulator
  float* out = (float*)d_out;

  // 1) zero accumulator (d_ws is poisoned; replays must not see stale sums)
  const long long n_agg = (long long)batch * 16;
  int zgrid = (int)((n_agg + 255) / 256);
  zero_f32<<<zgrid, 256, 0, stream>>>(agg, n_agg);

  // 2) edge gather + atomic scatter-add (bandwidth-dominant phase)
  const long long ethreads = n_edges * 4;
  int egrid = (int)((ethreads + 255) / 256);
  edge_scatter<<<egrid, 256, 0, stream>>>(x, esrc, edst, agg, n_edges);

  // 3) fused SAGE linear + 2-layer MLP via f32 WMMA, one wave per 16-row tile
  const int tiles = (batch + 15) / 16;
  int mgrid = (tiles + 3) / 4;           // 4 waves (128 threads) per block
  sage_mlp_wmma<<<mgrid, 128, 0, stream>>>(agg, x, Wl, bl, Wr,
                                           W1, b1, W2, b2, out, batch);
}